// QCQPLoss_55903294324986
// MI455X (gfx1250) — compile-verified
//
#include <hip/hip_runtime.h>
#include <hip/hip_bf16.h>

#define THREADS 256

// ---------------------------------------------------------------------------
// Kernel 1: per-element 4x4 symmetric eigensolve (cyclic Jacobi) + loss,
// block partial sums. A_vec staged to LDS via CDNA5 async global->LDS DMA
// (GLOBAL_LOAD_ASYNC_TO_LDS_B128, ASYNCcnt) -- 40B/element AoS becomes fully
// coalesced b128 traffic; per-lane LDS gather is bank-conflict-free.
// ---------------------------------------------------------------------------
__global__ __launch_bounds__(THREADS) void qcqp_main_kernel(
    const float* __restrict__ A_vec,      // (B,10)
    const float4* __restrict__ q_t,       // (B,4) as float4
    float* __restrict__ partial,          // (numBlocks)
    int B)
{
    __shared__ __align__(16) float2 sA2[THREADS * 5];   // 10240 B staging tile
    __shared__ float swave[THREADS / 32];

    const int tid   = threadIdx.x;
    const int base  = blockIdx.x * THREADS;
    const int nelem = min(THREADS, B - base);
    const int nf4   = (nelem * 10 + 3) >> 2;    // float4 count to stage (<=640)

    // uniform 64-bit global base for this block's tile (16B aligned: 10240|16)
    const unsigned long long gbase =
        (unsigned long long)(A_vec) + (unsigned long long)base * 40ull;
    // low 32 bits of the flat shared address == LDS byte offset (aperture in [63:32])
    const unsigned ldsbase = (unsigned)(unsigned long long)&sA2[0];

    // Async global->LDS staging: coalesced b128 per lane, tracked by ASYNCcnt.
    #pragma unroll
    for (int it = 0; it < 3; ++it) {
        int i4 = it * THREADS + tid;
        if (i4 < nf4) {
            unsigned lds  = ldsbase + (unsigned)(i4 * 16);
            unsigned voff = (unsigned)(i4 * 16);
            asm volatile("global_load_async_to_lds_b128 %0, %1, %2"
                         :
                         : "v"(lds), "v"(voff), "s"(gbase)
                         : "memory");
        }
    }

    // Overlap: pull the (already coalesced) target quaternion while DMA runs.
    const int elem = base + tid;
    float4 qt4 = make_float4(0.f, 0.f, 0.f, 0.f);
    if (elem < B) qt4 = q_t[elem];

    asm volatile("s_wait_asynccnt 0" ::: "memory");
    __syncthreads();

    // Gather this lane's 10 coefficients as 5x b64 (8B-aligned stride of 40B;
    // banks (10t+2j) mod 64 are distinct across a wave32 -> conflict-free).
    float v[10];
    #pragma unroll
    for (int j = 0; j < 5; ++j) {
        float2 p = sA2[tid * 5 + j];
        v[2 * j]     = -p.x;
        v[2 * j + 1] = -p.y;
    }

    // Symmetric 4x4 from upper triangle (row-major triu order).
    float a[4][4];
    a[0][0]=v[0]; a[0][1]=v[1]; a[0][2]=v[2]; a[0][3]=v[3];
    a[1][0]=v[1]; a[1][1]=v[4]; a[1][2]=v[5]; a[1][3]=v[6];
    a[2][0]=v[2]; a[2][1]=v[5]; a[2][2]=v[7]; a[2][3]=v[8];
    a[3][0]=v[3]; a[3][1]=v[6]; a[3][2]=v[8]; a[3][3]=v[9];

    float V[4][4] = {{1,0,0,0},{0,1,0,0},{0,0,1,0},{0,0,0,1}};

    const int PP[6] = {0,0,0,1,1,2};
    const int QQ[6] = {1,2,3,2,3,3};

    // Cyclic Jacobi: 5 sweeps x 6 rotations. (p,q) constant per unrolled
    // rotation so a[][] and V[][] stay fully in VGPRs. Rotation parameters
    // use v_rcp/v_rsq approximations: Jacobi is self-correcting, final-sweep
    // parameter error ~1e-7 relative, far below fp32 eigenvector tolerance.
    #pragma unroll 1
    for (int sw = 0; sw < 5; ++sw) {
        #pragma unroll
        for (int r = 0; r < 6; ++r) {
            const int p = PP[r], q = QQ[r];
            const float apq = a[p][q];
            const float app = a[p][p], aqq = a[q][q];
            float t, c, s;
            if (__builtin_fabsf(apq) > 1e-20f) {
                float theta = (aqq - app) * 0.5f * __frcp_rn(apq);
                float denom = __builtin_fabsf(theta) +
                              __builtin_sqrtf(__builtin_fmaf(theta, theta, 1.0f));
                t = __builtin_copysignf(__frcp_rn(denom), theta);
                c = __frsqrt_rn(__builtin_fmaf(t, t, 1.0f));
                s = t * c;
            } else { t = 0.f; c = 1.f; s = 0.f; }

            a[p][p] = app - t * apq;
            a[q][q] = aqq + t * apq;
            a[p][q] = 0.f; a[q][p] = 0.f;
            #pragma unroll
            for (int i = 0; i < 4; ++i) {
                if (i == p || i == q) continue;
                const float aip = a[i][p], aiq = a[i][q];
                const float nip = c * aip - s * aiq;
                const float niq = s * aip + c * aiq;
                a[i][p] = nip; a[p][i] = nip;
                a[i][q] = niq; a[q][i] = niq;
            }
            #pragma unroll
            for (int i = 0; i < 4; ++i) {
                const float vip = V[i][p], viq = V[i][q];
                V[i][p] = c * vip - s * viq;
                V[i][q] = s * vip + c * viq;
            }
        }
    }

    // Column of minimum eigenvalue (diagonal after convergence).
    const float d0 = a[0][0], d1 = a[1][1], d2e = a[2][2], d3 = a[3][3];
    const int   m01 = (d1 < d0) ? 1 : 0;
    const float e01 = fminf(d0, d1);
    const int   m23 = (d3 < d2e) ? 3 : 2;
    const float e23 = fminf(d2e, d3);
    const int   m   = (e23 < e01) ? m23 : m01;

    float qp[4];
    #pragma unroll
    for (int i = 0; i < 4; ++i)
        qp[i] = (m == 0) ? V[i][0] : (m == 1) ? V[i][1] : (m == 2) ? V[i][2] : V[i][3];

    const float qt[4] = {qt4.x, qt4.y, qt4.z, qt4.w};
    float dm = 0.f, dp = 0.f;
    #pragma unroll
    for (int i = 0; i < 4; ++i) {
        const float u = qp[i] - qt[i];
        const float w = qp[i] + qt[i];
        dm += u * u;
        dp += w * w;
    }
    const float d2 = fminf(dm, dp);
    float loss = 2.0f * d2 * (4.0f - d2);
    if (elem >= B) loss = 0.f;

    // wave32 tree reduction, then per-block partial (deterministic order).
    #pragma unroll
    for (int off = 16; off > 0; off >>= 1)
        loss += __shfl_down(loss, off, 32);

    const int wid = tid >> 5, lane = tid & 31;
    if (lane == 0) swave[wid] = loss;
    __syncthreads();
    if (tid == 0) {
        float ssum = 0.f;
        #pragma unroll
        for (int w = 0; w < THREADS / 32; ++w) ssum += swave[w];
        partial[blockIdx.x] = ssum;
    }
}

// ---------------------------------------------------------------------------
// Kernel 2: deterministic final reduction of block partials -> mean.
// ---------------------------------------------------------------------------
__global__ __launch_bounds__(THREADS) void qcqp_reduce_kernel(
    const float* __restrict__ partial, int n, float invB, float* __restrict__ out)
{
    __shared__ float sm[THREADS];
    float s = 0.f;
    for (int i = threadIdx.x; i < n; i += THREADS) s += partial[i];
    sm[threadIdx.x] = s;
    __syncthreads();
    #pragma unroll
    for (int off = THREADS / 2; off > 0; off >>= 1) {
        if (threadIdx.x < off) sm[threadIdx.x] += sm[threadIdx.x + off];
        __syncthreads();
    }
    if (threadIdx.x == 0) out[0] = sm[0] * invB;
}

extern "C" void kernel_launch(void* const* d_in, const int* in_sizes, int n_in,
                              void* d_out, int out_size, void* d_ws, size_t ws_size,
                              hipStream_t stream) {
    const float*  A_vec = (const float*)d_in[0];
    const float4* q_t   = (const float4*)d_in[1];

    const int B  = in_sizes[0] / 10;
    const int nb = (B + THREADS - 1) / THREADS;

    float* partial = (float*)d_ws;

    qcqp_main_kernel<<<nb, THREADS, 0, stream>>>(A_vec, q_t, partial, B);
    qcqp_reduce_kernel<<<1, THREADS, 0, stream>>>(partial, nb, 1.0f / (float)B,
                                                  (float*)d_out);
}